// GCN_18382460027177
// MI455X (gfx1250) — compile-verified
//
#include <hip/hip_runtime.h>
#include <hip/hip_bf16.h>

typedef __attribute__((ext_vector_type(2))) float v2f;
typedef __attribute__((ext_vector_type(8))) float v8f;

#define IN_CH  128
#define HID_CH 64

// ---------------------------------------------------------------------------
// Utility: fill a float buffer with a constant
// ---------------------------------------------------------------------------
__global__ __launch_bounds__(256)
void gcn_fill_kernel(float* __restrict__ p, float v, long long n) {
    long long i = (long long)blockIdx.x * blockDim.x + threadIdx.x;
    if (i < n) p[i] = v;
}

// ---------------------------------------------------------------------------
// Degree accumulation: deg[col[e]] += 1   (deg pre-initialized to 1.0 = self loop)
// ---------------------------------------------------------------------------
__global__ __launch_bounds__(256)
void gcn_deg_kernel(const int* __restrict__ col, float* __restrict__ deg, int nE) {
    int e = blockIdx.x * blockDim.x + threadIdx.x;
    if (e < nE) atomicAdd(deg + col[e], 1.0f);
}

// ---------------------------------------------------------------------------
// In-place deg -> rsqrt(deg)
// ---------------------------------------------------------------------------
__global__ __launch_bounds__(256)
void gcn_rsqrt_kernel(float* __restrict__ deg, int n) {
    int i = blockIdx.x * blockDim.x + threadIdx.x;
    if (i < n) {
        float d = deg[i];
        deg[i] = (d > 0.0f) ? rsqrtf(d) : 0.0f;
    }
}

// ---------------------------------------------------------------------------
// H = X @ W using V_WMMA_F32_16X16X4_F32.
// Block = 128 threads = 4 waves; each wave owns one 16x16 output tile
// (16 rows shared by the block, 16 of the 64 output channels per wave).
// A-frag  (16x4 f32, 2 VGPR): lane l -> m = l&15, K = kb + 2*(l>>4) + v
// B-frag  ( 4x16 f32, 2 VGPR): lane l -> n = l&15, K = kb + 2*(l>>4) + v
// C/D     (16x16 f32, 8 VGPR): (m = v + 8*(l>>4), n = l&15)
// ---------------------------------------------------------------------------
__global__ __launch_bounds__(128)
void gcn_gemm_wmma_kernel(const float* __restrict__ X, const float* __restrict__ W,
                          float* __restrict__ H, int K, int nNodes) {
    const int lane = threadIdx.x & 31;
    const int wave = threadIdx.x >> 5;      // 0..3 : 16-channel tile
    const int n    = lane & 15;             // column within tile (and A row m)
    const int half = lane >> 4;             // lane-half selects K pair / m offset
    const int r0   = blockIdx.x * 16;       // 16-row tile
    const int n0   = wave * 16;

    // clamp A-row reads so partial tail tiles never read OOB (N=100000 is exact anyway)
    const int rA = (r0 + n < nNodes) ? (r0 + n) : (nNodes - 1);

    v8f acc = {};
    for (int kb = 0; kb < K; kb += 4) {
        const int ka = kb + 2 * half;                       // even -> 8B aligned
        v2f a = *(const v2f*)(X + (long long)rA * K + ka);  // A[m][ka], A[m][ka+1]
        v2f b;
        b.x = W[(long long)(ka    ) * HID_CH + n0 + n];     // B[ka  ][n]
        b.y = W[(long long)(ka + 1) * HID_CH + n0 + n];     // B[ka+1][n]
        acc = __builtin_amdgcn_wmma_f32_16x16x4_f32(
                  /*neg_a=*/false, a, /*neg_b=*/false, b,
                  /*c_mod=*/(short)0, acc, /*reuse_a=*/false, /*reuse_b=*/false);
    }

#pragma unroll
    for (int v = 0; v < 8; ++v) {
        int m = v + 8 * half;
        if (r0 + m < nNodes)
            H[(long long)(r0 + m) * HID_CH + n0 + n] = acc[v];
    }
}

// ---------------------------------------------------------------------------
// Edge scatter: out[col[e]] += h[row[e]] * dinv[row[e]] * dinv[col[e]]
// One wave32 per edge, 2 channels per lane (64 channels).
// Gather is one coalesced 256B row read; accumulate via global_atomic_add_f32.
// ---------------------------------------------------------------------------
__global__ __launch_bounds__(256)
void gcn_scatter_kernel(const float* __restrict__ H, const int* __restrict__ row,
                        const int* __restrict__ col, const float* __restrict__ dinv,
                        float* __restrict__ out, int nE) {
    const int lane = threadIdx.x & 31;
    const int e    = blockIdx.x * 8 + (threadIdx.x >> 5);
    if (e >= nE) return;

    const int   r   = row[e];
    const int   c   = col[e];
    const float nrm = dinv[r] * dinv[c];

    const float* hr = H + (long long)r * HID_CH;
    float v0 = hr[lane];
    float v1 = hr[lane + 32];

    float* oc = out + (long long)c * HID_CH;
    atomicAdd(oc + lane,      v0 * nrm);
    atomicAdd(oc + lane + 32, v1 * nrm);
}

// ---------------------------------------------------------------------------
// Finalize: out = agg + h * dinv^2 (self-loop term) + bias  [+ ReLU]
// ---------------------------------------------------------------------------
__global__ __launch_bounds__(256)
void gcn_finalize_kernel(const float* __restrict__ agg, const float* __restrict__ h,
                         const float* __restrict__ dinv, const float* __restrict__ bias,
                         float* __restrict__ out, int nNodes, int doRelu) {
    long long idx = (long long)blockIdx.x * blockDim.x + threadIdx.x;
    if (idx >= (long long)nNodes * HID_CH) return;
    int node = (int)(idx >> 6);
    int ch   = (int)(idx & 63);
    float di = dinv[node];
    float v  = agg[idx] + h[idx] * di * di + bias[ch];
    out[idx] = doRelu ? fmaxf(v, 0.0f) : v;
}

// ---------------------------------------------------------------------------
extern "C" void kernel_launch(void* const* d_in, const int* in_sizes, int n_in,
                              void* d_out, int out_size, void* d_ws, size_t ws_size,
                              hipStream_t stream) {
    const float* x   = (const float*)d_in[0];
    const int*   ei  = (const int*)  d_in[1];   // edge_index [2, E] flat
    const float* W1  = (const float*)d_in[2];
    const float* b1  = (const float*)d_in[3];
    const float* W2  = (const float*)d_in[4];
    const float* b2  = (const float*)d_in[5];
    float*       out = (float*)d_out;

    const int N  = in_sizes[0] / IN_CH;         // 100000
    const int nE = in_sizes[1] / 2;             // 1600000
    const int*   rowp = ei;
    const int*   colp = ei + nE;

    // workspace layout
    float* deg = (float*)d_ws;                  // N        (becomes dinv in place)
    float* h1  = deg + N;                       // N*64
    float* agg = h1 + (long long)N * HID_CH;    // N*64
    float* h2  = agg + (long long)N * HID_CH;   // N*64

    const long long feat = (long long)N * HID_CH;
    const int fillBlocks = (int)((feat + 255) / 256);
    const int edgeThreadBlocks = (nE + 255) / 256;
    const int edgeWaveBlocks   = (nE + 7) / 8;  // 8 edges (waves) per 256-thr block
    const int nodeBlocks = (N + 255) / 256;
    const int tileBlocks = (N + 15) / 16;

    // init: deg = 1 (self loop), agg1 = 0, out (= agg2 target) = 0
    gcn_fill_kernel<<<(N + 255) / 256, 256, 0, stream>>>(deg, 1.0f, N);
    gcn_fill_kernel<<<fillBlocks, 256, 0, stream>>>(agg, 0.0f, feat);
    gcn_fill_kernel<<<fillBlocks, 256, 0, stream>>>(out, 0.0f, feat);

    // normalization coefficients
    gcn_deg_kernel<<<edgeThreadBlocks, 256, 0, stream>>>(colp, deg, nE);
    gcn_rsqrt_kernel<<<nodeBlocks, 256, 0, stream>>>(deg, N);

    // ---- layer 1 ----
    gcn_gemm_wmma_kernel<<<tileBlocks, 128, 0, stream>>>(x, W1, h1, IN_CH, N);
    gcn_scatter_kernel<<<edgeWaveBlocks, 256, 0, stream>>>(h1, rowp, colp, deg, agg, nE);
    gcn_finalize_kernel<<<fillBlocks, 256, 0, stream>>>(agg, h1, deg, b1, h1, N, /*relu=*/1);

    // ---- layer 2 ----
    gcn_gemm_wmma_kernel<<<tileBlocks, 128, 0, stream>>>(h1, W2, h2, HID_CH, N);
    gcn_scatter_kernel<<<edgeWaveBlocks, 256, 0, stream>>>(h2, rowp, colp, deg, out, nE);
    gcn_finalize_kernel<<<fillBlocks, 256, 0, stream>>>(out, h2, deg, b2, out, N, /*relu=*/0);
}